// AttributePredictor_25658134626769
// MI455X (gfx1250) — compile-verified
//
#include <hip/hip_runtime.h>
#include <hip/hip_bf16.h>
#include <cstddef>

typedef __attribute__((ext_vector_type(2))) float v2f;
typedef __attribute__((ext_vector_type(8))) float v8f;

#define C_CH 512
#define NT 1024
#define HH 32
#define WW 32
#define OO 7
#define KBOX 256
#define RDIM (C_CH * OO * OO)   // 25088
#define NOUT 512
#define X_ELEMS (8 * NT * C_CH) // 4194304

// ---------------------------------------------------------------- kernel 1: passthrough copy of x
__global__ __launch_bounds__(256) void copy_x_kernel(const float4* __restrict__ in,
                                                     float4* __restrict__ out, int n4) {
    int i = blockIdx.x * blockDim.x + threadIdx.x;
    if (i < n4) out[i] = in[i];
}

// ---------------------------------------------------------------- kernel 2: ROI max pool
// grid = (49, 256): blockIdx.x = bin ij, blockIdx.y = box k. 256 threads stride channels.
// pool layout: pool[k][c*49 + ij]  (matches conv_w's [o][c*49+ij] reduction layout)
__global__ __launch_bounds__(256) void roipool_kernel(const float* __restrict__ x,
                                                      const float* __restrict__ boxes,
                                                      float* __restrict__ pool) {
    const int ij = blockIdx.x;          // 0..48
    const int k  = blockIdx.y;          // 0..255
    const int i  = ij / OO;
    const int j  = ij % OO;

    const float* bx = boxes + (size_t)k * 5;
    const int   b  = (int)bx[0];
    const float x1 = rintf(bx[1] * 32.0f);
    const float y1 = rintf(bx[2] * 32.0f);
    const float x2 = rintf(bx[3] * 32.0f);
    const float y2 = rintf(bx[4] * 32.0f);
    const float rw = fmaxf(x2 - x1 + 1.0f, 1.0f);
    const float rh = fmaxf(y2 - y1 + 1.0f, 1.0f);

    const float hlo = fminf(fmaxf(floorf(((float)i)       * rh / 7.0f) + y1, 0.0f), 32.0f);
    const float hhi = fminf(fmaxf(ceilf (((float)(i + 1)) * rh / 7.0f) + y1, 0.0f), 32.0f);
    const float wlo = fminf(fmaxf(floorf(((float)j)       * rw / 7.0f) + x1, 0.0f), 32.0f);
    const float whi = fminf(fmaxf(ceilf (((float)(j + 1)) * rw / 7.0f) + x1, 0.0f), 32.0f);

    const int hs = (int)hlo, he = (int)hhi;
    const int ws = (int)wlo, we = (int)whi;
    const bool nonempty = (he > hs) && (we > ws);

    const float* xb = x + (size_t)b * NT * C_CH;
    for (int c = threadIdx.x; c < C_CH; c += 256) {
        float m = -3.402823466e38f;
        for (int h = hs; h < he; ++h) {
            const float* row = xb + ((size_t)(h * WW) * C_CH) + c;
            for (int w = ws; w < we; ++w) {
                m = fmaxf(m, row[(size_t)w * C_CH]);
            }
        }
        pool[(size_t)k * RDIM + (size_t)c * (OO * OO) + ij] = nonempty ? m : 0.0f;
    }
}

// ---------------------------------------------------------------- kernel 3: WMMA f32 GEMM
// box_feats[m, n] = sum_r pool[m, r] * conv_w[n, r] + bias[n]
// M=256, N=512, R=25088. One 16x16 tile per wave, 8 waves/block.
// grid = (16 M-tiles, 4 N-groups), block = 256.
__global__ __launch_bounds__(256) void gemm_wmma_kernel(const float* __restrict__ pool,
                                                        const float* __restrict__ w,
                                                        const float* __restrict__ bias,
                                                        float* __restrict__ out) {
    const int wave = threadIdx.x >> 5;
    const int lane = threadIdx.x & 31;
    const int m0 = blockIdx.x * 16;
    const int n0 = (blockIdx.y * 8 + wave) * 16;
    const int hi = lane >> 4;        // 0: K pair {r, r+1}; 1: K pair {r+2, r+3}
    const int l  = lane & 15;        // A: row M=l   B: col N=l

    const float* aRow = pool + (size_t)(m0 + l) * RDIM + (size_t)(hi * 2);
    const float* bCol = w    + (size_t)(n0 + l) * RDIM + (size_t)(hi * 2);

    v8f acc0 = {}; v8f acc1 = {}; v8f acc2 = {}; v8f acc3 = {};

    for (int r = 0; r < RDIM; r += 16) {   // 1568 iterations, 4 independent acc chains
        v2f a0 = *(const v2f*)(aRow + r);
        v2f b0 = *(const v2f*)(bCol + r);
        v2f a1 = *(const v2f*)(aRow + r + 4);
        v2f b1 = *(const v2f*)(bCol + r + 4);
        v2f a2 = *(const v2f*)(aRow + r + 8);
        v2f b2 = *(const v2f*)(bCol + r + 8);
        v2f a3 = *(const v2f*)(aRow + r + 12);
        v2f b3 = *(const v2f*)(bCol + r + 12);
        acc0 = __builtin_amdgcn_wmma_f32_16x16x4_f32(false, a0, false, b0, (short)0, acc0, false, false);
        acc1 = __builtin_amdgcn_wmma_f32_16x16x4_f32(false, a1, false, b1, (short)0, acc1, false, false);
        acc2 = __builtin_amdgcn_wmma_f32_16x16x4_f32(false, a2, false, b2, (short)0, acc2, false, false);
        acc3 = __builtin_amdgcn_wmma_f32_16x16x4_f32(false, a3, false, b3, (short)0, acc3, false, false);
    }

    v8f acc = (acc0 + acc1) + (acc2 + acc3);

    const int n = n0 + l;
    const float bv = bias[n];
#pragma unroll
    for (int v = 0; v < 8; ++v) {
        const int m = m0 + v + hi * 8;   // D layout: VGPR v holds M=v (lanes 0-15), M=v+8 (lanes 16-31)
        out[(size_t)m * NOUT + n] = acc[v] + bv;
    }
}

// ---------------------------------------------------------------- launcher
extern "C" void kernel_launch(void* const* d_in, const int* in_sizes, int n_in,
                              void* d_out, int out_size, void* d_ws, size_t ws_size,
                              hipStream_t stream) {
    const float* x      = (const float*)d_in[0];   // (8, 1024, 512)
    const float* boxes  = (const float*)d_in[1];   // (256, 5)
    // d_in[2] = box_labels (unused)
    const float* conv_w = (const float*)d_in[3];   // (512, 512, 7, 7)
    const float* conv_b = (const float*)d_in[4];   // (512,)

    float* out_x     = (float*)d_out;              // first 4194304 floats: x passthrough
    float* out_feats = (float*)d_out + X_ELEMS;    // next 256*512 floats: box_feats
    float* pool_ws   = (float*)d_ws;               // 256 * 25088 floats = ~25.7 MB

    // 1) passthrough copy of x
    {
        const int n4 = X_ELEMS / 4;
        copy_x_kernel<<<dim3((n4 + 255) / 256), dim3(256), 0, stream>>>(
            (const float4*)x, (float4*)out_x, n4);
    }

    // 2) ROI max pool -> workspace
    {
        roipool_kernel<<<dim3(OO * OO, KBOX), dim3(256), 0, stream>>>(x, boxes, pool_ws);
    }

    // 3) WMMA GEMM: (256 x 25088) * (25088 x 512)^T + bias
    {
        gemm_wmma_kernel<<<dim3(KBOX / 16, NOUT / (8 * 16)), dim3(256), 0, stream>>>(
            pool_ws, conv_w, conv_b, out_feats);
    }
}